// GBST_42434276884940
// MI455X (gfx1250) — compile-verified
//
#include <hip/hip_runtime.h>
#include <math.h>

// GBST-style fused kernel for MI455X (gfx1250, wave32).
//
// out[b, u, h] = sum_{t in {2u,2u+1}} 0.5 * sum_m s[t][m] * mean_{t' in win_m(t)} x[t'][h]
//             = (Wd @ x_chunk)[u][h]   with Wd a small banded matrix per 96-pos chunk.
// The GEMM runs on V_WMMA_F32_16X16X4_F32 (full f32 precision).

typedef __attribute__((ext_vector_type(2))) float v2f;
typedef __attribute__((ext_vector_type(8))) float v8f;

#define SEQ    3072
#define EMBED  256
#define BATCH  16
#define TCH    96      // chunk length; multiple of lcm(1,2,3,4, DOWNSAMPLE)=12
#define OUTCH  48      // output rows per chunk (TCH / 2)
#define XS     272     // padded LDS row stride for x (272*4B: +32-bank shift between K-halves)
#define WDS    100     // padded LDS row stride for Wd (36 mod 64 -> conflict-free A rows)

// ---------------------------------------------------------------------------
// Kernel 1: one-time precompute (runs every launch; deterministic).
//   peg[4][256]  : sinusoidal positional table (period MAX_BLOCK=4)
//   ewg[256]     : embed_table[v] . w_score   (per-vocab score projection)
//   pwg[4]       : pe[p] . w_score
// ---------------------------------------------------------------------------
__global__ __launch_bounds__(256) void gbst_prep_kernel(
    const float* __restrict__ embed, const float* __restrict__ w,
    float* __restrict__ peg, float* __restrict__ ewg, float* __restrict__ pwg) {
  __shared__ float pesh[4 * EMBED];
  const int f = threadIdx.x;                       // 0..255 (feature)
  const int i = f >> 1;
  const float div = expf((float)(2 * i) * (-9.210340371976184f / 256.0f)); // 10000^(-i/128)
  #pragma unroll
  for (int p = 0; p < 4; ++p) {
    const float ang = (float)p * div;
    const float v = (f & 1) ? cosf(ang) : sinf(ang);
    pesh[p * EMBED + f] = v;
    peg[p * EMBED + f] = v;
  }
  // ewg: one vocab row per thread
  float acc = 0.0f;
  for (int k = 0; k < EMBED; ++k) acc += embed[f * EMBED + k] * w[k];
  ewg[f] = acc;
  __syncthreads();
  if (f < 4) {
    float a = 0.0f;
    for (int k = 0; k < EMBED; ++k) a += pesh[f * EMBED + k] * w[k];
    pwg[f] = a;
  }
}

// ---------------------------------------------------------------------------
// Kernel 2: fused main pass. One workgroup = one (batch, 96-position chunk).
// ---------------------------------------------------------------------------
__global__ __launch_bounds__(256) void gbst_main_kernel(
    const int*   __restrict__ ids,
    const float* __restrict__ embed,
    const float* __restrict__ peg,
    const float* __restrict__ ewg,
    const float* __restrict__ pwg,
    float*       __restrict__ out) {
  extern __shared__ float lds[];          // [TCH][XS] x-chunk, then [OUTCH][WDS] Wd
  float* xsh = lds;
  float* wd  = lds + TCH * XS;
  __shared__ int   ids_s[TCH];
  __shared__ float ysh[TCH];              // per-position scalar score x_t . w
  __shared__ float ssh[TCH * 4];          // calibrated block weights s[t][m]

  const int b   = blockIdx.y;
  const int c   = blockIdx.x;
  const int t0  = c * TCH;                // multiple of 96 -> divisible by 12
  const int tid = threadIdx.x;

  // ---- Phase 0: ids + scalar base scores ---------------------------------
  if (tid < TCH) {
    const int id = ids[b * SEQ + t0 + tid];
    ids_s[tid] = id;
    ysh[tid]   = ewg[id] + pwg[tid & 3];  // (t0+t)%4 == t%4 since 4 | t0
  }
  __syncthreads();

  // ---- Phase 1: gather x chunk into LDS (emb[id] + pe[t%4]) --------------
  for (int t = 0; t < TCH; ++t) {
    const int id = ids_s[t];
    xsh[t * XS + tid] = embed[id * EMBED + tid] + peg[(t & 3) * EMBED + tid];
  }

  // ---- Phase 2: block-size softmax + 4x4 score calibration (per position)
  if (tid < TCH) {
    const float sc0 = ysh[tid];
    const int t2 = tid & ~1;
    const float sc1 = 0.5f * (ysh[t2] + ysh[t2 + 1]);
    const int t3 = tid - tid % 3;
    const float sc2 = (ysh[t3] + ysh[t3 + 1] + ysh[t3 + 2]) * (1.0f / 3.0f);
    const int t4 = tid & ~3;
    const float sc3 = 0.25f * (ysh[t4] + ysh[t4 + 1] + ysh[t4 + 2] + ysh[t4 + 3]);
    // softmax over the 4 block sizes (b_score cancels -> omitted)
    const float mx = fmaxf(fmaxf(sc0, sc1), fmaxf(sc2, sc3));
    const float e0 = expf(sc0 - mx), e1 = expf(sc1 - mx);
    const float e2 = expf(sc2 - mx), e3 = expf(sc3 - mx);
    const float inv = 1.0f / (e0 + e1 + e2 + e3);
    float sv[4] = { e0 * inv, e1 * inv, e2 * inv, e3 * inv };
    // attn = softmax_rows(s s^T); s = attn @ s   (all sv > 0)
    const float sm = fmaxf(fmaxf(sv[0], sv[1]), fmaxf(sv[2], sv[3]));
    #pragma unroll
    for (int i = 0; i < 4; ++i) {
      float num = 0.0f, den = 0.0f;
      #pragma unroll
      for (int j = 0; j < 4; ++j) {
        const float e = expf(sv[i] * (sv[j] - sm));
        num += e * sv[j];
        den += e;
      }
      ssh[tid * 4 + i] = num / den;
    }
  }
  __syncthreads();

  // ---- Phase 3: build banded downsampled weight matrix Wd[48][96] --------
  // Wd[u][t'] = sum_{t in {2u,2u+1}} sum_m (0.5 * s[t][m] / bs) * [t' in win_bs(t)]
  if (tid < OUTCH) {
    float* row = &wd[tid * WDS];
    for (int k = 0; k < TCH; ++k) row[k] = 0.0f;
    #pragma unroll
    for (int h = 0; h < 2; ++h) {
      const int t = 2 * tid + h;
      #pragma unroll
      for (int m = 0; m < 4; ++m) {
        const int bs = m + 1;
        const int base = (t / bs) * bs;          // chunk-local: bs | t0
        const float wv = 0.5f * ssh[t * 4 + m] / (float)bs;
        for (int k = 0; k < bs; ++k) row[base + k] += wv;
      }
    }
  }
  __syncthreads();

  // ---- Phase 4: out_chunk[48][256] = Wd[48][96] @ x[96][256] on WMMA -----
  // f32 WMMA fragment layout (ISA 7.12.2):
  //   A 16x4 : lanes 0-15 -> {K0,K1} in v0,v1 ; lanes 16-31 -> {K2,K3}
  //   B 4x16 : lanes 0-15 -> rows K0,K1 at col=lane ; lanes 16-31 -> K2,K3
  //   D      : VGPR r -> row r (lanes 0-15) / row 8+r (lanes 16-31)
  const int lane = tid & 31;
  const int wave = tid >> 5;
  const int l2   = lane & 15;
  const int kh   = (lane >> 4) << 1;             // 0 or 2
  #pragma unroll 1
  for (int i = 0; i < 6; ++i) {                  // 8 waves x 6 tiles = 48 tiles
    const int tile = wave * 6 + i;
    const int mt = tile >> 4;                    // 0..2  (M tile)
    const int nt = tile & 15;                    // 0..15 (N tile)
    v8f acc = {0.f, 0.f, 0.f, 0.f, 0.f, 0.f, 0.f, 0.f};
    const float* arow = &wd[(mt * 16 + l2) * WDS];
    const float* bcol = &xsh[nt * 16 + l2];
    for (int ks = 0; ks < TCH / 4; ++ks) {       // 24 K-steps
      const int k0 = ks * 4 + kh;
      v2f a;  a.x  = arow[k0];        a.y  = arow[k0 + 1];       // ds_load_b64
      v2f bb; bb.x = bcol[k0 * XS];   bb.y = bcol[(k0 + 1) * XS];
      acc = __builtin_amdgcn_wmma_f32_16x16x4_f32(
          /*neg_a=*/false, a, /*neg_b=*/false, bb,
          /*c_mod=*/(short)0, acc, /*reuse_a=*/false, /*reuse_b=*/false);
    }
    const int ug  = c * OUTCH + mt * 16 + ((lane >> 4) << 3);
    const int col = nt * 16 + l2;
    float* op = out + ((size_t)b * (SEQ / 2) + ug) * EMBED + col;
    #pragma unroll
    for (int r = 0; r < 8; ++r) op[(size_t)r * EMBED] = acc[r];
  }
}

// ---------------------------------------------------------------------------
extern "C" void kernel_launch(void* const* d_in, const int* in_sizes, int n_in,
                              void* d_out, int out_size, void* d_ws, size_t ws_size,
                              hipStream_t stream) {
  (void)in_sizes; (void)n_in; (void)out_size; (void)ws_size;
  const int*   ids   = (const int*)d_in[0];     // [16,3072]
  const float* embed = (const float*)d_in[1];   // [256,256]
  const float* w     = (const float*)d_in[2];   // [1,256]
  // d_in[3] (b_score) cancels inside the block-size softmax -> unused.

  float* ws  = (float*)d_ws;
  float* peg = ws;              // 4*256
  float* ewg = ws + 1024;       // 256
  float* pwg = ws + 1280;       // 4

  gbst_prep_kernel<<<dim3(1), dim3(256), 0, stream>>>(embed, w, peg, ewg, pwg);

  const size_t shbytes = (size_t)(TCH * XS + OUTCH * WDS) * sizeof(float); // 123648 B
  dim3 grid(SEQ / TCH, BATCH);                                             // 32 x 16
  gbst_main_kernel<<<grid, dim3(256), shbytes, stream>>>(
      ids, embed, peg, ewg, pwg, (float*)d_out);
}